// BresenhamAttention_57286273794753
// MI455X (gfx1250) — compile-verified
//
#include <hip/hip_runtime.h>

typedef __attribute__((ext_vector_type(2))) float v2f;
typedef __attribute__((ext_vector_type(4))) float v4f;
typedef __attribute__((ext_vector_type(8))) float v8f;

#define CCH   256      // channels
#define HWPX  50176    // 224*224 flattened spatial, contiguous per (b,c)
#define TILE  128      // pixels per workgroup tile
#define WIDTH 224

// Fused: att = sigmoid(sum_c x*w + b); *2 on border; out = x * att.
// One global read of x, one global write of out (memory-bound optimum:
// 1.64 GB total traffic -> ~70 us floor at 23.3 TB/s).
__global__ __launch_bounds__(256) void border_att_fused(
    const float* __restrict__ x, const float* __restrict__ conv_w,
    const float* __restrict__ conv_b, float* __restrict__ out)
{
    extern __shared__ float smem[];
    float* xtile = smem;                 // [CCH][TILE] fp32 = 128 KB
    float* wsm   = smem + CCH * TILE;    // [CCH]
    float* comb  = wsm + CCH;            // [TILE]

    const int tid = threadIdx.x;
    const int blocksPerImg = HWPX / TILE;              // 392
    const int b       = blockIdx.x / blocksPerImg;
    const int pixbase = (blockIdx.x % blocksPerImg) * TILE;

    // stage weights (256 floats)
    wsm[tid] = conv_w[tid];

    // ---- Phase 1: global -> LDS, b128 per lane, 8 channels per pass ----
    const int cg = tid >> 5;     // channel group 0..7
    const int ln = tid & 31;     // lane
    for (int it = 0; it < CCH / 8; ++it) {
        const int c = it * 8 + cg;
        const size_t goff = ((size_t)b * CCH + c) * HWPX + pixbase + ln * 4;
        v4f v = __builtin_nontemporal_load((const v4f*)(x + goff));
        *(v4f*)(xtile + c * TILE + ln * 4) = v;
    }
    __syncthreads();

    // ---- Phase 2: channel reduction via V_WMMA_F32_16X16X4_F32 ----
    // A[m][k] = w[c+k] (broadcast over rows m), B[k][n] = xtile[c+k][pix0+n]
    // => D[m][n] = att contribution for pixel pix0+n, identical over m.
    // f32 A 16x4 layout: lanes 0-15 hold K0(V0),K1(V1); lanes 16-31 K2,K3.
    const int wave = tid >> 5;
    const int lane = tid & 31;
    const int half = lane >> 4;          // 0: K{0,1}, 1: K{2,3}
    const int n    = lane & 15;          // column / pixel-in-16
    const int pix0 = wave * 16;          // 8 waves * 16 px = 128 px tile

    v8f acc = {};
    #pragma unroll 4
    for (int c = 0; c < CCH; c += 4) {
        v2f a, bm;
        a.x  = wsm[c + 2 * half + 0];
        a.y  = wsm[c + 2 * half + 1];
        bm.x = xtile[(c + 2 * half + 0) * TILE + pix0 + n];
        bm.y = xtile[(c + 2 * half + 1) * TILE + pix0 + n];
        acc = __builtin_amdgcn_wmma_f32_16x16x4_f32(
            /*neg_a=*/false, a, /*neg_b=*/false, bm,
            /*c_mod=*/(short)0, acc, /*reuse_a=*/false, /*reuse_b=*/false);
    }

    // D row M=0 lives in acc[0]: lanes 0-15 -> (M=0,N=lane), so each of
    // lanes 0..15 holds att for pixel pix0+lane.
    const float bias = conv_b[0];
    if (lane < 16) {
        const float att = acc[0] + bias;
        // sigmoid via v_exp_f32 + v_rcp_f32 (avoids IEEE div sequence)
        float s = __builtin_amdgcn_rcpf(1.0f + __expf(-att));
        const int pglob = pixbase + pix0 + lane;         // 0..50175
        const int h = pglob / WIDTH;
        const int w = pglob - h * WIDTH;
        if ((h == 0) | (h == WIDTH - 1) | (w == 0) | (w == WIDTH - 1))
            s *= 2.0f;                                    // att*(1+mask)
        comb[pix0 + lane] = s;
    }
    __syncthreads();

    // ---- Phase 3: LDS -> global, scale by comb, b128 stores ----
    const v4f cv = *(const v4f*)(comb + ln * 4);         // loop-invariant
    for (int it = 0; it < CCH / 8; ++it) {
        const int c = it * 8 + cg;
        v4f v = *(const v4f*)(xtile + c * TILE + ln * 4);
        v *= cv;
        const size_t goff = ((size_t)b * CCH + c) * HWPX + pixbase + ln * 4;
        __builtin_nontemporal_store(v, (v4f*)(out + goff));
    }
}

extern "C" void kernel_launch(void* const* d_in, const int* in_sizes, int n_in,
                              void* d_out, int out_size, void* d_ws, size_t ws_size,
                              hipStream_t stream) {
    const float* x  = (const float*)d_in[0];   // (16,256,224,224) fp32
    const float* cw = (const float*)d_in[1];   // (1,256,1,1) fp32
    const float* cb = (const float*)d_in[2];   // (1,) fp32
    float* out = (float*)d_out;

    const int grid = 16 * (HWPX / TILE);       // 6272 workgroups
    const size_t shmem = (size_t)(CCH * TILE + CCH + TILE) * sizeof(float); // ~129.5 KB

    // opt in to >64KB dynamic LDS (no-op if already allowed); not a stream op
    (void)hipFuncSetAttribute((const void*)border_att_fused,
                              hipFuncAttributeMaxDynamicSharedMemorySize, (int)shmem);

    border_att_fused<<<grid, 256, shmem, stream>>>(x, cw, cb, out);
}